// CIN_33019708572224
// MI455X (gfx1250) — compile-verified
//
#include <hip/hip_runtime.h>

typedef float v2f __attribute__((ext_vector_type(2)));
typedef float v4f __attribute__((ext_vector_type(4)));
typedef float v8f __attribute__((ext_vector_type(8)));

#define B_DIM 8
#define C_DIM 256
#define S_DIM 16
#define HW_DIM 65536   // 256*256
#define EPS 1e-5f

// ---------------------------------------------------------------------------
// Kernel 1: gamma_mix / beta_mix = style_weights[8,16] @ {gammas,betas}[16,256]
// via V_WMMA_F32_16X16X4_F32, one wave32. A padded to 16 rows (rows 8..15 = 0).
// A layout (16x4 f32): lanes 0-15 hold K=k0,k0+1 in vgpr0/1; lanes 16-31 hold
// K=k0+2,k0+3. B (4x16): row K striped across lanes within a VGPR, same K
// split across lane halves. D (16x16 f32): vgpr v, lane L -> M = v + 8*(L/16),
// N = L%16; only M<8 is real output.
// ---------------------------------------------------------------------------
__global__ __launch_bounds__(32) void cin_mix_wmma(
    const float* __restrict__ sw,      // [8,16]
    const float* __restrict__ gammas,  // [16,256]
    const float* __restrict__ betas,   // [16,256]
    float* __restrict__ mix)           // [2, 8, 256] (gamma_mix, beta_mix)
{
    const int lane = threadIdx.x;
    const int half = lane >> 4;   // 0: lanes 0-15, 1: lanes 16-31
    const int row  = lane & 15;   // M for A, N-within-tile for B/D

    // Preload A fragments for the 4 K-steps (k0 = 0,4,8,12).
    float a0[4], a1[4];
#pragma unroll
    for (int kk = 0; kk < 4; ++kk) {
        const int ka = kk * 4 + 2 * half;
        float x0 = 0.0f, x1 = 0.0f;
        if (row < B_DIM) {
            x0 = sw[row * S_DIM + ka];
            x1 = sw[row * S_DIM + ka + 1];
        }
        a0[kk] = x0;
        a1[kk] = x1;
    }

    for (int nt = 0; nt < C_DIM / 16; ++nt) {
        const int col = nt * 16 + row;
        v8f dg = {};
        v8f db = {};
#pragma unroll
        for (int kk = 0; kk < 4; ++kk) {
            const int ka = kk * 4 + 2 * half;
            v2f a;  a.x  = a0[kk];                  a.y  = a1[kk];
            v2f bg; bg.x = gammas[ka * C_DIM + col]; bg.y = gammas[(ka + 1) * C_DIM + col];
            v2f bb; bb.x = betas [ka * C_DIM + col]; bb.y = betas [(ka + 1) * C_DIM + col];
            dg = __builtin_amdgcn_wmma_f32_16x16x4_f32(false, a, false, bg,
                                                       (short)0, dg, false, false);
            db = __builtin_amdgcn_wmma_f32_16x16x4_f32(false, a, false, bb,
                                                       (short)0, db, false, false);
        }
        // D rows M = v + 8*half; batch rows live in half==0, v = 0..7.
        if (half == 0) {
#pragma unroll
            for (int v = 0; v < 8; ++v) {
                mix[v * C_DIM + col]                 = dg[v];  // gamma_mix[b=v][c=col]
                mix[B_DIM * C_DIM + v * C_DIM + col] = db[v];  // beta_mix
            }
        }
    }
}

// ---------------------------------------------------------------------------
// Kernel 2: one block per (b,c) plane. 1024 threads x 16 floats held in
// registers -> exactly one HBM read + one HBM write per element (roofline
// floor ~46us at 23.3 TB/s). Non-temporal: pure streaming, zero reuse.
// ---------------------------------------------------------------------------
__global__ __launch_bounds__(1024) void cin_norm(
    const float* __restrict__ x,
    const float* __restrict__ mix,
    float* __restrict__ out)
{
    const int bc = blockIdx.x;            // b*C + c, 0..2047
    const int t  = threadIdx.x;           // 0..1023
    const size_t base4 = (size_t)bc * (HW_DIM / 4);
    const v4f* __restrict__ xp = reinterpret_cast<const v4f*>(x) + base4;
    v4f* __restrict__ op       = reinterpret_cast<v4f*>(out) + base4;

    v4f r[4];
    float s1 = 0.0f, s2 = 0.0f;
#pragma unroll
    for (int w = 0; w < 4; ++w) {
        r[w] = __builtin_nontemporal_load(&xp[w * 1024 + t]);
        s1 += (r[w].x + r[w].y) + (r[w].z + r[w].w);
        s2 = fmaf(r[w].x, r[w].x, s2);
        s2 = fmaf(r[w].y, r[w].y, s2);
        s2 = fmaf(r[w].z, r[w].z, s2);
        s2 = fmaf(r[w].w, r[w].w, s2);
    }

    // wave32 butterfly reduction
#pragma unroll
    for (int off = 16; off > 0; off >>= 1) {
        s1 += __shfl_xor(s1, off, 32);
        s2 += __shfl_xor(s2, off, 32);
    }

    __shared__ float ls1[32];
    __shared__ float ls2[32];
    __shared__ float sb[2];
    const int wave = t >> 5;
    const int lane = t & 31;
    if (lane == 0) { ls1[wave] = s1; ls2[wave] = s2; }
    __syncthreads();

    if (wave == 0) {
        float a = ls1[lane];
        float b = ls2[lane];
#pragma unroll
        for (int off = 16; off > 0; off >>= 1) {
            a += __shfl_xor(a, off, 32);
            b += __shfl_xor(b, off, 32);
        }
        if (lane == 0) {
            const float inv_n = 1.0f / (float)HW_DIM;
            const float mean  = a * inv_n;
            const float var   = fmaf(-mean, mean, b * inv_n);  // biased var
            const float rstd  = rsqrtf(var + EPS);
            const float gm    = mix[bc];                       // gamma_mix[b][c]
            const float bm    = mix[B_DIM * C_DIM + bc];       // beta_mix[b][c]
            const float scale = gm * rstd;
            sb[0] = scale;
            sb[1] = fmaf(-mean, scale, bm);                    // beta - mean*scale
        }
    }
    __syncthreads();

    const float scale = sb[0];
    const float shift = sb[1];
#pragma unroll
    for (int w = 0; w < 4; ++w) {
        v4f o;
        o.x = fmaf(r[w].x, scale, shift);
        o.y = fmaf(r[w].y, scale, shift);
        o.z = fmaf(r[w].z, scale, shift);
        o.w = fmaf(r[w].w, scale, shift);
        __builtin_nontemporal_store(o, &op[w * 1024 + t]);
    }
}

extern "C" void kernel_launch(void* const* d_in, const int* in_sizes, int n_in,
                              void* d_out, int out_size, void* d_ws, size_t ws_size,
                              hipStream_t stream)
{
    const float* x      = (const float*)d_in[0];  // [8,256,256,256]
    const float* sw     = (const float*)d_in[1];  // [8,16]
    const float* gammas = (const float*)d_in[2];  // [16,256]
    const float* betas  = (const float*)d_in[3];  // [16,256]
    float* out = (float*)d_out;
    float* mix = (float*)d_ws;                    // 2*8*256 floats = 16 KB

    cin_mix_wmma<<<1, 32, 0, stream>>>(sw, gammas, betas, mix);
    cin_norm<<<B_DIM * C_DIM, 1024, 0, stream>>>(x, mix, out);
}